// MaskedLoss_87522843558384
// MI455X (gfx1250) — compile-verified
//
#include <hip/hip_runtime.h>
#include <math.h>

// Problem constants (from reference): B=8, S=512, V=32000
#define Bsz 8
#define Ssz 512
#define Vsz 32000
#define NROWS (Bsz * Ssz)          // 4096 rows
#define BDIM 320                   // 10 wave32 waves per row
#define TILE_FLOATS (BDIM * 4)     // 1280 floats = 5120 B per tile (b128 per lane)
#define NTILES (Vsz / TILE_FLOATS) // 25 tiles exactly
#define DEPTH 4                    // async pipeline depth (LDS ring buffers)

#if defined(__gfx1250__) && __has_builtin(__builtin_amdgcn_global_load_async_to_lds_b128)
#define USE_ASYNC 1
#else
#define USE_ASYNC 0
#endif

#if USE_ASYNC
// Builtin parameter types (confirmed by compiler diagnostics): 16-byte int
// vector pointers, global (AS1) source and LDS (AS3) destination.
typedef int v4i __attribute__((__vector_size__(4 * sizeof(int))));
typedef __attribute__((address_space(1))) v4i gv4i_t;
typedef __attribute__((address_space(3))) v4i lv4i_t;

__device__ __forceinline__ void issue_tile(const float* rowp, int tile,
                                           float* bufbase, int tid) {
    gv4i_t* g = (gv4i_t*)(rowp + (size_t)tile * TILE_FLOATS) + tid;
    lv4i_t* l = (lv4i_t*)bufbase + tid;
    __builtin_amdgcn_global_load_async_to_lds_b128(g, l, 0, 0);
}

// s_wait_asynccnt needs a literal immediate; async ops complete in order per
// wave, so waiting "<= n" guarantees the oldest of n+1 outstanding is done.
__device__ __forceinline__ void wait_async_upto(int n) {
    switch (n) {
    case 0:  asm volatile("s_wait_asynccnt 0x0" ::: "memory"); break;
    case 1:  asm volatile("s_wait_asynccnt 0x1" ::: "memory"); break;
    case 2:  asm volatile("s_wait_asynccnt 0x2" ::: "memory"); break;
    default: asm volatile("s_wait_asynccnt 0x3" ::: "memory"); break;
    }
}
#endif

// Online logsumexp single-element update: track running max m and sum s of exp(x - m).
__device__ __forceinline__ void lse_add(float& m, float& s, float x) {
    if (x > m) {
        s = s * __expf(m - x) + 1.0f;
        m = x;
    } else {
        s += __expf(x - m);
    }
}

// Combine two (m, s) partial logsumexp states.
__device__ __forceinline__ void lse_combine(float& m, float& s, float mo, float so) {
    float M = fmaxf(m, mo);
    s = s * __expf(m - M) + so * __expf(mo - M);
    m = M;
}

__global__ __launch_bounds__(BDIM) void row_nll_kernel(
    const float* __restrict__ predict,   // (B*S, V)
    const int* __restrict__ target,      // (B*S)
    float* __restrict__ tok_nll)         // (B*S) workspace output
{
    const int row = blockIdx.x;
    const int tid = threadIdx.x;
    const float* rowp = predict + (size_t)row * Vsz;

    float m = -3.402823466e38f; // -FLT_MAX
    float s = 0.0f;

#if USE_ASYNC
    // 4-deep ring of LDS tiles fed by the async global->LDS engine (ASYNCcnt).
    // Each lane consumes exactly the 16 B it loads, so NO barriers are needed
    // in the streaming loop: waves run fully decoupled, each gated only by its
    // own s_wait_asynccnt. Intra-wave buffer reuse is safe because a tile's
    // ds_load value is consumed (DScnt-waited) before the overwriting issue.
    __shared__ __align__(16) float buf[DEPTH][TILE_FLOATS];

    #pragma unroll
    for (int p = 0; p < DEPTH - 1; ++p)      // prime tiles 0..DEPTH-2
        issue_tile(rowp, p, &buf[p][0], tid);

    #pragma unroll 1
    for (int t = 0; t < NTILES; ++t) {
        const int nxt = t + DEPTH - 1;
        if (nxt < NTILES)
            issue_tile(rowp, nxt, &buf[nxt & (DEPTH - 1)][0], tid);
        // Outstanding after optional issue: min(DEPTH, NTILES - t).
        const int rem = NTILES - 1 - t;
        wait_async_upto(rem < (DEPTH - 1) ? rem : (DEPTH - 1));

        float4 x = ((const float4*)&buf[t & (DEPTH - 1)][0])[tid];
        lse_add(m, s, x.x);
        lse_add(m, s, x.y);
        lse_add(m, s, x.z);
        lse_add(m, s, x.w);
    }
#else
    // Fallback: direct b128 streaming loads.
    for (int t = 0; t < NTILES; ++t) {
        float4 x = ((const float4*)rowp)[t * BDIM + tid];
        lse_add(m, s, x.x);
        lse_add(m, s, x.y);
        lse_add(m, s, x.z);
        lse_add(m, s, x.w);
    }
#endif

    // Wave32 shuffle reduction of (m, s).
    for (int off = 16; off > 0; off >>= 1) {
        float mo = __shfl_down(m, off, 32);
        float so = __shfl_down(s, off, 32);
        lse_combine(m, s, mo, so);
    }

    __shared__ float wm[BDIM / 32];
    __shared__ float wsum[BDIM / 32];
    const int wave = tid >> 5;
    const int lane = tid & 31;
    if (lane == 0) { wm[wave] = m; wsum[wave] = s; }
    __syncthreads();   // the only workgroup barrier in this kernel

    if (tid == 0) {
        float M = wm[0], S = wsum[0];
        #pragma unroll
        for (int w = 1; w < BDIM / 32; ++w) lse_combine(M, S, wm[w], wsum[w]);
        int tg = target[row];
        float xt = rowp[tg];
        // -log_softmax(x)[tg] = logsumexp(x) - x[tg]
        tok_nll[row] = (logf(S) + M) - xt;
    }
}

__global__ __launch_bounds__(256) void finalize_kernel(
    const float* __restrict__ tok_nll,   // (B*S)
    const int* __restrict__ mask,        // (B,1) -> word_len per example
    float* __restrict__ out)             // scalar
{
    __shared__ float red[256];
    const int tid = threadIdx.x;
    float total = 0.0f;

    for (int b = 0; b < Bsz; ++b) {
        int wl = mask[b];
        float acc = 0.0f;
        for (int si = tid; si < Ssz; si += 256)
            if (si < wl) acc += tok_nll[b * Ssz + si];
        red[tid] = acc;
        __syncthreads();
        for (int off = 128; off > 0; off >>= 1) {
            if (tid < off) red[tid] += red[tid + off];
            __syncthreads();
        }
        if (tid == 0) total += red[0] / (float)wl;
        __syncthreads();
    }
    if (tid == 0) out[0] = total / (float)Bsz;
}

extern "C" void kernel_launch(void* const* d_in, const int* in_sizes, int n_in,
                              void* d_out, int out_size, void* d_ws, size_t ws_size,
                              hipStream_t stream) {
    const float* predict = (const float*)d_in[0];
    const int*   target  = (const int*)d_in[1];   // int64 in reference -> int32 on device per harness
    const int*   mask    = (const int*)d_in[2];
    float* out = (float*)d_out;
    float* tok = (float*)d_ws;                    // B*S floats = 16 KB scratch

    row_nll_kernel<<<NROWS, BDIM, 0, stream>>>(predict, target, tok);
    finalize_kernel<<<1, 256, 0, stream>>>(tok, mask, out);
}